// TestModule_65085934404074
// MI455X (gfx1250) — compile-verified
//
#include <hip/hip_runtime.h>
#include <math.h>

#define N_NODES 20000
#define DIM 256
#define N_EDGES 320000
#define LDS_STRIDE 260   // 260 mod 64 == 4 -> conflict-free strided A-fragment reads

typedef float v2f __attribute__((ext_vector_type(2)));
typedef float v8f __attribute__((ext_vector_type(8)));

// Native L2-side f32 atomic add, device scope (agg is shared across workgroups;
// WGP$ is read-only so the RMW executes in GL2's atomic units).
__device__ __forceinline__ void atomic_add_f32_dev(float* p, float v) {
    asm volatile("global_atomic_add_f32 %0, %1, off scope:SCOPE_DEV"
                 :: "v"(p), "v"(v) : "memory");
}

// 32-bit LDS byte offset (AS(3) pointer value) for async-to-LDS destinations.
__device__ __forceinline__ unsigned lds_offset_of(const void* p) {
    return (unsigned)(size_t)(const __attribute__((address_space(3))) void*)p;
}

// ---------------------------------------------------------------------------
// zero a float buffer (n4 = count of float4's)
// ---------------------------------------------------------------------------
__global__ void k_zero4(float4* __restrict__ p, long n4) {
    long i = (long)blockIdx.x * blockDim.x + threadIdx.x;
    long stride = (long)gridDim.x * blockDim.x;
    float4 z = make_float4(0.f, 0.f, 0.f, 0.f);
    for (; i < n4; i += stride) p[i] = z;
}

// ---------------------------------------------------------------------------
// scatter-add: agg[dst[e]] += x[src[e]]   (64 lanes per edge, float4 chunks)
// ---------------------------------------------------------------------------
__global__ void k_scatter(const float* __restrict__ x,
                          const int* __restrict__ src,
                          const int* __restrict__ dst,
                          float* __restrict__ agg) {
    int tid = blockIdx.x * blockDim.x + threadIdx.x;
    int e = tid >> 6;            // 64 threads per edge
    if (e >= N_EDGES) return;
    int c = (tid & 63) << 2;     // float column 0..252
    int s = src[e];
    int d = dst[e];
    const float4 v = *reinterpret_cast<const float4*>(x + (long)s * DIM + c);
    float* o = agg + (long)d * DIM + c;
    atomic_add_f32_dev(o + 0, v.x);
    atomic_add_f32_dev(o + 1, v.y);
    atomic_add_f32_dev(o + 2, v.z);
    atomic_add_f32_dev(o + 3, v.w);
}

// ---------------------------------------------------------------------------
// fused dual GEMM accumulate:
//   C[m0:m0+16, :] += A1_strip @ B1 + A2_strip @ B2 + bias
// 512 threads = 16 waves; wave w computes the 16x16 tile at columns [16w,16w+16)
// using V_WMMA_F32_16X16X4_F32 over K=256 (64 steps x 2 matrices).
// A strips staged global->LDS via the async copy engine (ASYNCcnt).
// ---------------------------------------------------------------------------
__global__ __launch_bounds__(512)
void k_gemm_fused(const float* __restrict__ A1, const float* __restrict__ B1,
                  const float* __restrict__ A2, const float* __restrict__ B2,
                  const float* __restrict__ bias, float* __restrict__ C) {
    __shared__ float lds[2][16 * LDS_STRIDE];

    const int m0 = blockIdx.x * 16;
    const int t  = threadIdx.x;

    // stage both A strips (16 x 256 floats) into padded LDS rows with
    // global_load_async_to_lds_b128: per lane, 16B global -> 16B LDS.
    for (int s = 0; s < 2; ++s) {
        const float* A = s ? A2 : A1;
        for (int i = t; i < 16 * 64; i += 512) {           // 1024 float4 per strip
            int row = i >> 6;
            int c4  = (i & 63) << 2;
            unsigned goff  = (unsigned)(((m0 + row) * DIM + c4) * 4);  // bytes
            unsigned laddr = lds_offset_of(&lds[s][row * LDS_STRIDE + c4]);
            asm volatile("global_load_async_to_lds_b128 %0, %1, %2"
                         :: "v"(laddr), "v"(goff), "s"(A) : "memory");
        }
    }
    asm volatile("s_wait_asynccnt 0x0" ::: "memory");
    __syncthreads();

    const int wave  = t >> 5;           // 0..15 -> N tile
    const int lane  = t & 31;
    const int n0    = wave * 16;
    const int ml    = lane & 15;        // A: row M; B: col N within tile
    const int khalf = (lane >> 4) << 1; // lanes 0-15: K+0/K+1, lanes 16-31: K+2/K+3
    const int col   = n0 + ml;

    // aligned float2 views of the A rows in LDS (row byte offset 1040 % 8 == 0)
    const float2* a1p = reinterpret_cast<const float2*>(&lds[0][ml * LDS_STRIDE]);
    const float2* a2p = reinterpret_cast<const float2*>(&lds[1][ml * LDS_STRIDE]);

    v8f acc = {};
    for (int k0 = 0; k0 < DIM; k0 += 4) {
        const int kb = k0 + khalf;               // even
        float2 a1 = a1p[kb >> 1];
        float2 a2 = a2p[kb >> 1];
        v2f af1 = {a1.x, a1.y};
        v2f af2 = {a2.x, a2.y};
        v2f bf1 = {B1[(long)kb * DIM + col], B1[(long)(kb + 1) * DIM + col]};
        v2f bf2 = {B2[(long)kb * DIM + col], B2[(long)(kb + 1) * DIM + col]};
        acc = __builtin_amdgcn_wmma_f32_16x16x4_f32(false, af1, false, bf1,
                                                    (short)0, acc, false, false);
        acc = __builtin_amdgcn_wmma_f32_16x16x4_f32(false, af2, false, bf2,
                                                    (short)0, acc, false, false);
    }

    // C/D layout: VGPR j, lanes 0-15 -> M=j, lanes 16-31 -> M=j+8; N = lane&15
    const float b = bias[col];
    const int rbase = (lane < 16) ? 0 : 8;
    #pragma unroll
    for (int j = 0; j < 8; ++j) {
        long idx = (long)(m0 + rbase + j) * DIM + col;
        C[idx] += acc[j] + b;
    }
}

// ---------------------------------------------------------------------------
// in-place ReLU (n4 float4's)
// ---------------------------------------------------------------------------
__global__ void k_relu4(float4* __restrict__ p, long n4) {
    long i = (long)blockIdx.x * blockDim.x + threadIdx.x;
    long stride = (long)gridDim.x * blockDim.x;
    for (; i < n4; i += stride) {
        float4 v = p[i];
        v.x = fmaxf(v.x, 0.f); v.y = fmaxf(v.y, 0.f);
        v.z = fmaxf(v.z, 0.f); v.w = fmaxf(v.w, 0.f);
        p[i] = v;
    }
}

// ---------------------------------------------------------------------------
// per-row NLL: one wave (32 lanes) per row of 256 logits; 8 rows per block.
// partial[blockIdx] = sum of 8 row NLLs (deterministic order).
// ---------------------------------------------------------------------------
__global__ __launch_bounds__(256)
void k_loss_partial(const float* __restrict__ logits, const int* __restrict__ y,
                    float* __restrict__ partial) {
    __shared__ float row_nll[8];
    const int wave = threadIdx.x >> 5;
    const int lane = threadIdx.x & 31;
    const int row  = blockIdx.x * 8 + wave;
    const float* z = logits + (long)row * DIM;

    float vals[8];
    float m = -INFINITY;
    #pragma unroll
    for (int i = 0; i < 8; ++i) {
        vals[i] = z[lane + 32 * i];
        m = fmaxf(m, vals[i]);
    }
    for (int off = 16; off; off >>= 1) m = fmaxf(m, __shfl_xor(m, off, 32));
    float sum = 0.f;
    #pragma unroll
    for (int i = 0; i < 8; ++i) sum += __expf(vals[i] - m);
    for (int off = 16; off; off >>= 1) sum += __shfl_xor(sum, off, 32);

    if (lane == 0) {
        float lse = __logf(sum) + m;
        row_nll[wave] = lse - z[y[row]];
    }
    __syncthreads();
    if (threadIdx.x == 0) {
        float s = 0.f;
        #pragma unroll
        for (int i = 0; i < 8; ++i) s += row_nll[i];
        partial[blockIdx.x] = s;
    }
}

__global__ __launch_bounds__(256)
void k_final_reduce(const float* __restrict__ partial, int nb,
                    float* __restrict__ out, float invN) {
    __shared__ float s[256];
    float acc = 0.f;
    for (int i = threadIdx.x; i < nb; i += 256) acc += partial[i];
    s[threadIdx.x] = acc;
    __syncthreads();
    for (int off = 128; off; off >>= 1) {
        if (threadIdx.x < off) s[threadIdx.x] += s[threadIdx.x + off];
        __syncthreads();
    }
    if (threadIdx.x == 0) out[0] = s[0] * invN;
}

// ---------------------------------------------------------------------------
// host-side launch
// ---------------------------------------------------------------------------
extern "C" void kernel_launch(void* const* d_in, const int* in_sizes, int n_in,
                              void* d_out, int out_size, void* d_ws, size_t ws_size,
                              hipStream_t stream) {
    (void)in_sizes; (void)n_in; (void)out_size; (void)ws_size;

    const float* x       = (const float*)d_in[0];   // [3, N, D]
    const int*   edges   = (const int*)  d_in[1];   // [6, 2, E]
    const int*   y       = (const int*)  d_in[2];   // [N]
    const float* W_rel1  = (const float*)d_in[3];   // [6, D, D]
    const float* b_rel1  = (const float*)d_in[4];   // [6, D]
    const float* W_root1 = (const float*)d_in[5];   // [6, D, D]
    const float* W_rel2  = (const float*)d_in[6];   // [6, D, D]
    const float* b_rel2  = (const float*)d_in[7];   // [6, D]
    const float* W_root2 = (const float*)d_in[8];   // [6, D, D]
    float* out = (float*)d_out;

    const long ND  = (long)N_NODES * DIM;           // 5,120,000 floats
    float* h1      = (float*)d_ws;                  // [3, N, D]
    float* h2      = h1 + 3 * ND;                   // [N, D] (dst type 0 only)
    float* agg     = h2 + ND;                       // [N, D] scratch
    float* partial = agg + ND;                      // [2500]

    static const int REL_SRC[6] = {0, 0, 1, 1, 2, 2};
    static const int REL_DST[6] = {0, 1, 0, 2, 1, 2};

    const int ZB = 1024;                                       // zero/relu threads per block
    const long h1_4 = 3 * ND / 4, nd_4 = ND / 4;
    const int scatter_blocks = (N_EDGES * 64) / 256;           // 80000
    const int gemm_blocks = N_NODES / 16;                      // 1250

    // ---- layer 1: h1 = relu( sum_k  agg_k@W_rel1[k] + x[t]@W_root1[k] + b ) ----
    k_zero4<<<4096, ZB, 0, stream>>>((float4*)h1, h1_4);
    for (int k = 0; k < 6; ++k) {
        const int s = REL_SRC[k], t = REL_DST[k];
        const int* src = edges + (long)k * 2 * N_EDGES;
        const int* dst = src + N_EDGES;
        k_zero4<<<4096, ZB, 0, stream>>>((float4*)agg, nd_4);
        k_scatter<<<scatter_blocks, 256, 0, stream>>>(x + s * ND, src, dst, agg);
        k_gemm_fused<<<gemm_blocks, 512, 0, stream>>>(
            agg, W_rel1 + (long)k * DIM * DIM,
            x + t * ND, W_root1 + (long)k * DIM * DIM,
            b_rel1 + (long)k * DIM, h1 + t * ND);
    }
    k_relu4<<<4096, ZB, 0, stream>>>((float4*)h1, h1_4);

    // ---- layer 2: only dst type 0 is needed -> relations k=0 (0->0), k=2 (1->0) ----
    k_zero4<<<4096, ZB, 0, stream>>>((float4*)h2, nd_4);
    {
        const int ks[2] = {0, 2};
        for (int i = 0; i < 2; ++i) {
            const int k = ks[i];
            const int s = REL_SRC[k];
            const int* src = edges + (long)k * 2 * N_EDGES;
            const int* dst = src + N_EDGES;
            k_zero4<<<4096, ZB, 0, stream>>>((float4*)agg, nd_4);
            k_scatter<<<scatter_blocks, 256, 0, stream>>>(h1 + s * ND, src, dst, agg);
            k_gemm_fused<<<gemm_blocks, 512, 0, stream>>>(
                agg, W_rel2 + (long)k * DIM * DIM,
                h1 /* root input = h1[0] */, W_root2 + (long)k * DIM * DIM,
                b_rel2 + (long)k * DIM, h2);
        }
    }

    // ---- loss: mean NLL of log_softmax(h2) over 256 classes ----
    const int loss_blocks = N_NODES / 8;   // 2500
    k_loss_partial<<<loss_blocks, 256, 0, stream>>>(h2, y, partial);
    k_final_reduce<<<1, 256, 0, stream>>>(partial, loss_blocks, out, 1.0f / N_NODES);
}